// GCN_70119636075235
// MI455X (gfx1250) — compile-verified
//
#include <hip/hip_runtime.h>
#include <hip/hip_bf16.h>
#include <math.h>

#define NN    100000
#define F_IN  512
#define HD    128
#define NC    40
#define NL    2
#define BN_EPS 1e-5f

typedef __attribute__((ext_vector_type(2))) float v2f;
typedef __attribute__((ext_vector_type(8))) float v8f;

// ---------------------------------------------------------------------------
// Hardware f32 atomic add (CDNA5 GLOBAL_ATOMIC_ADD_F32, no return -> STOREcnt).
// Device scope: destinations are shared across workgroups; plain accumulation
// needs no ordering beyond the kernel boundary.
// ---------------------------------------------------------------------------
__device__ __forceinline__ void atomic_add_f32_dev(float* p, float v) {
    asm volatile("global_atomic_add_f32 %0, %1, off scope:SCOPE_DEV"
                 :: "v"(p), "v"(v)
                 : "memory");
}

// ---------------------------------------------------------------------------
// Degree / normalization
// ---------------------------------------------------------------------------
__global__ void deg_init_kernel(float* __restrict__ deg, int n) {
    int i = blockIdx.x * blockDim.x + threadIdx.x;
    if (i < n) deg[i] = 1.0f;                       // self-loop weight
}

__global__ void deg_scatter_kernel(const int* __restrict__ col,
                                   const float* __restrict__ w,
                                   float* __restrict__ deg, int e) {
    int i = blockIdx.x * blockDim.x + threadIdx.x;
    if (i < e) atomic_add_f32_dev(&deg[col[i]], w[i]);
}

__global__ void dinv_kernel(const float* __restrict__ deg,
                            float* __restrict__ dinv, int n) {
    int i = blockIdx.x * blockDim.x + threadIdx.x;
    if (i < n) {
        float d = deg[i];
        dinv[i] = (d > 0.0f) ? rsqrtf(fmaxf(d, 1e-12f)) : 0.0f;
    }
}

// ---------------------------------------------------------------------------
// Pack B (KxNcols row-major) into WMMA f32 16x16x4 B-fragment order:
//   P[(kstep*NT + ntile)*32 + lane] = { B[k][n], B[k+1][n] }
//   k = 4*kstep + ((lane>=16)?2:0),  n = 16*ntile + (lane&15)  (zero-pad n)
// ---------------------------------------------------------------------------
__global__ void pack_b_kernel(const float* __restrict__ B, float* __restrict__ P,
                              int K, int ncols, int ntiles) {
    int tid   = blockIdx.x * blockDim.x + threadIdx.x;
    int total = (K / 4) * ntiles * 32;
    if (tid >= total) return;
    int lane  = tid & 31;
    int nt    = (tid >> 5) % ntiles;
    int kstep = tid / (32 * ntiles);
    int k     = kstep * 4 + ((lane >> 4) << 1);
    int n     = nt * 16 + (lane & 15);
    v2f v;
    v.x = (n < ncols) ? B[(size_t)k * ncols + n]       : 0.0f;
    v.y = (n < ncols) ? B[(size_t)(k + 1) * ncols + n] : 0.0f;
    ((v2f*)P)[tid] = v;
}

__global__ void pad_bout_kernel(const float* __restrict__ b, float* __restrict__ bp) {
    int i = threadIdx.x;
    if (i < 48) bp[i] = (i < NC) ? b[i] : 0.0f;
}

// ---------------------------------------------------------------------------
// WMMA GEMM: D[M x (NT*16)] = A[M x K] * Bpacked, one wave per 16-row M tile.
// MODE 0: store raw.  MODE 1: bias+ReLU+BatchNorm.  MODE 2: bias only.
// ---------------------------------------------------------------------------
template <int NT, int MODE>
__launch_bounds__(256)
__global__ void gemm_wmma_kernel(const float* __restrict__ A,
                                 const float* __restrict__ Bp,
                                 float* __restrict__ D,
                                 int M, int K, int ldd,
                                 const float* __restrict__ bias,
                                 const float* __restrict__ g,
                                 const float* __restrict__ be,
                                 const float* __restrict__ mu,
                                 const float* __restrict__ var) {
    const int lane  = threadIdx.x & 31;
    const int mtile = (blockIdx.x * blockDim.x + threadIdx.x) >> 5;
    if (mtile * 16 >= M) return;

    const int mrow = mtile * 16 + (lane & 15);      // A frag: M = lane&15
    const int koff = (lane >> 4) << 1;              // K split 0 / 2 per half-wave
    const float* __restrict__ arow = A + (size_t)mrow * K + koff;
    const v2f*  __restrict__ Bv    = (const v2f*)Bp;

    v8f acc[NT];
#pragma unroll
    for (int t = 0; t < NT; ++t) acc[t] = (v8f){0, 0, 0, 0, 0, 0, 0, 0};

    for (int ks = 0; ks < K / 4; ++ks) {
        v2f a = *(const v2f*)(arow + ks * 4);
        const v2f* bbase = Bv + ((size_t)ks * NT) * 32 + lane;
#pragma unroll
        for (int t = 0; t < NT; ++t) {
            v2f b = bbase[(size_t)t * 32];
            acc[t] = __builtin_amdgcn_wmma_f32_16x16x4_f32(
                false, a, false, b, (short)0, acc[t], false, false);
        }
    }

    // C/D layout: VGPR j -> M = j + 8*(lane>=16), N = lane&15
    const int mbase = mtile * 16 + ((lane >> 4) << 3);
#pragma unroll
    for (int t = 0; t < NT; ++t) {
        const int col = t * 16 + (lane & 15);
        float cb = 0.f, cg = 1.f, cbe = 0.f, cmu = 0.f, cis = 1.f;
        if (MODE >= 1) cb = bias[col];
        if (MODE == 1) {
            cg = g[col]; cbe = be[col]; cmu = mu[col];
            cis = rsqrtf(var[col] + BN_EPS);
        }
#pragma unroll
        for (int j = 0; j < 8; ++j) {
            float v = acc[t][j];
            if (MODE >= 1) v += cb;
            if (MODE == 1) { v = fmaxf(v, 0.0f); v = (v - cmu) * cis * cg + cbe; }
            D[(size_t)(mbase + j) * ldd + col] = v;
        }
    }
}

// ---------------------------------------------------------------------------
// Aggregation: agg = t * dinv^2 (self loop), then per-edge scatter-add in L2.
// ---------------------------------------------------------------------------
__global__ void agg_self_kernel(const float* __restrict__ t,
                                const float* __restrict__ dinv,
                                float* __restrict__ agg, int n) {
    int tid = blockIdx.x * blockDim.x + threadIdx.x;      // n * 32 threads
    if (tid >= n * 32) return;
    int node = tid >> 5, q = tid & 31;
    float s = dinv[node]; s = s * s;
    float4 v = ((const float4*)t)[(size_t)node * 32 + q];
    v.x *= s; v.y *= s; v.z *= s; v.w *= s;
    ((float4*)agg)[(size_t)node * 32 + q] = v;
}

__launch_bounds__(256)
__global__ void agg_edge_kernel(const int* __restrict__ row,
                                const int* __restrict__ col,
                                const float* __restrict__ w,
                                const float* __restrict__ dinv,
                                const float* __restrict__ t,
                                float* __restrict__ agg, int e) {
    int gid  = blockIdx.x * blockDim.x + threadIdx.x;
    int edge = gid >> 5;                                  // one wave per edge
    if (edge >= e) return;
    int lane = gid & 31;
    int r = row[edge], c = col[edge];
    float nrm = dinv[r] * w[edge] * dinv[c];
    float4 v = ((const float4*)(t + (size_t)r * HD))[lane];
    float* dst = agg + (size_t)c * HD + lane * 4;
    atomic_add_f32_dev(dst + 0, v.x * nrm);
    atomic_add_f32_dev(dst + 1, v.y * nrm);
    atomic_add_f32_dev(dst + 2, v.z * nrm);
    atomic_add_f32_dev(dst + 3, v.w * nrm);
}

__global__ void bnrelu_kernel(const float* __restrict__ agg,
                              const float* __restrict__ b,
                              const float* __restrict__ g,
                              const float* __restrict__ be,
                              const float* __restrict__ mu,
                              const float* __restrict__ var,
                              float* __restrict__ h, int n) {
    int tid = blockIdx.x * blockDim.x + threadIdx.x;      // n * HD threads
    if (tid >= n * HD) return;
    int f = tid & (HD - 1);
    float v = agg[tid] + b[f];
    v = fmaxf(v, 0.0f);
    h[tid] = (v - mu[f]) * rsqrtf(var[f] + BN_EPS) * g[f] + be[f];
}

__global__ void logsoftmax_kernel(const float* __restrict__ logits,
                                  float* __restrict__ out, int n) {
    int i = blockIdx.x * blockDim.x + threadIdx.x;
    if (i >= n) return;
    const float* lp = logits + (size_t)i * 48;
    float m = -INFINITY;
    for (int c = 0; c < NC; ++c) m = fmaxf(m, lp[c]);
    float s = 0.0f;
    for (int c = 0; c < NC; ++c) s += expf(lp[c] - m);
    float lse = m + logf(s);
    for (int c = 0; c < NC; ++c) out[(size_t)i * NC + c] = lp[c] - lse;
}

// ---------------------------------------------------------------------------
extern "C" void kernel_launch(void* const* d_in, const int* in_sizes, int n_in,
                              void* d_out, int out_size, void* d_ws, size_t ws_size,
                              hipStream_t stream) {
    const float* x        = (const float*)d_in[0];
    const int*   ei       = (const int*)d_in[1];
    const float* ew       = (const float*)d_in[2];
    const float* W_first  = (const float*)d_in[3];
    const float* b_first  = (const float*)d_in[4];
    const float* bn1_g    = (const float*)d_in[5];
    const float* bn1_b    = (const float*)d_in[6];
    const float* bn1_m    = (const float*)d_in[7];
    const float* bn1_v    = (const float*)d_in[8];
    const float* conv_W   = (const float*)d_in[9];
    const float* conv_b   = (const float*)d_in[10];
    const float* bn_g     = (const float*)d_in[11];
    const float* bn_b     = (const float*)d_in[12];
    const float* bn_m     = (const float*)d_in[13];
    const float* bn_v     = (const float*)d_in[14];
    const float* W_out    = (const float*)d_in[15];
    const float* b_out    = (const float*)d_in[16];
    float* out = (float*)d_out;

    const int N = in_sizes[0] / F_IN;        // 100000
    const int E = in_sizes[2];               // 1600000
    const int* erow = ei;
    const int* ecol = ei + E;

    // workspace layout (floats)
    float* ws    = (float*)d_ws;
    float* deg   = ws;                       // N
    float* dinv  = deg + N;                  // N
    float* h     = dinv + N;                 // N*HD
    float* t     = h + (size_t)N * HD;       // N*HD
    float* agg   = t + (size_t)N * HD;       // N*HD
    float* lgts  = agg + (size_t)N * HD;     // N*48
    float* wp1   = lgts + (size_t)N * 48;    // (512/4)*8*32*2 = 65536
    float* wp2a  = wp1 + 65536;              // (128/4)*8*32*2 = 16384
    float* wp2b  = wp2a + 16384;             // 16384
    float* wpo   = wp2b + 16384;             // (128/4)*3*32*2 = 6144
    float* bpad  = wpo + 6144;               // 48

    const int TB = 256;
    // --- GCN normalization ---
    deg_init_kernel<<<(N + TB - 1) / TB, TB, 0, stream>>>(deg, N);
    deg_scatter_kernel<<<(E + TB - 1) / TB, TB, 0, stream>>>(ecol, ew, deg, E);
    dinv_kernel<<<(N + TB - 1) / TB, TB, 0, stream>>>(deg, dinv, N);

    // --- weight repacks into WMMA B-fragment order ---
    pack_b_kernel<<<((F_IN / 4) * 8 * 32 + TB - 1) / TB, TB, 0, stream>>>(W_first, wp1, F_IN, HD, 8);
    pack_b_kernel<<<((HD / 4) * 8 * 32 + TB - 1) / TB, TB, 0, stream>>>(conv_W, wp2a, HD, HD, 8);
    pack_b_kernel<<<((HD / 4) * 8 * 32 + TB - 1) / TB, TB, 0, stream>>>(conv_W + HD * HD, wp2b, HD, HD, 8);
    pack_b_kernel<<<((HD / 4) * 3 * 32 + TB - 1) / TB, TB, 0, stream>>>(W_out, wpo, HD, NC, 3);
    pad_bout_kernel<<<1, 64, 0, stream>>>(b_out, bpad);

    const int mtiles = (N + 15) / 16;                    // 6250
    const int gblocks = (mtiles + 7) / 8;                // 8 waves / block

    // --- layer 1: h = BN(ReLU(x @ W_first + b_first)) ---
    gemm_wmma_kernel<8, 1><<<gblocks, TB, 0, stream>>>(
        x, wp1, h, N, F_IN, HD, b_first, bn1_g, bn1_b, bn1_m, bn1_v);

    // --- 2 GCNConv layers ---
    const float* wpk[NL] = {wp2a, wp2b};
    for (int i = 0; i < NL; ++i) {
        gemm_wmma_kernel<8, 0><<<gblocks, TB, 0, stream>>>(
            h, wpk[i], t, N, HD, HD, nullptr, nullptr, nullptr, nullptr, nullptr);
        agg_self_kernel<<<(N * 32 + TB - 1) / TB, TB, 0, stream>>>(t, dinv, agg, N);
        agg_edge_kernel<<<(int)(((size_t)E * 32 + TB - 1) / TB), TB, 0, stream>>>(
            erow, ecol, ew, dinv, t, agg, E);
        bnrelu_kernel<<<(N * HD + TB - 1) / TB, TB, 0, stream>>>(
            agg, conv_b + i * HD, bn_g + i * HD, bn_b + i * HD,
            bn_m + i * HD, bn_v + i * HD, h, N);
    }

    // --- head: logits (padded to 48 cols) + log_softmax ---
    gemm_wmma_kernel<3, 2><<<gblocks, TB, 0, stream>>>(
        h, wpo, lgts, N, HD, 48, bpad, nullptr, nullptr, nullptr, nullptr);
    logsoftmax_kernel<<<(N + TB - 1) / TB, TB, 0, stream>>>(lgts, out, N);
}